// RandomElasticDeformation3D_28192165331066
// MI455X (gfx1250) — compile-verified
//
#include <hip/hip_runtime.h>
#include <hip/hip_bf16.h>

// MI455X (gfx1250) fused random-elastic-deformation.
// Stage A (tiny): build composite 128x4 weight matrix Wtab = GaussSAME(16-tap) o CubicResizeWeights.
// Stage B (fused): per-wave, reconstruct the smooth flow for a 16x16 (h,w) tile at fixed (b,d)
//   via V_WMMA_F32_16X16X4_F32 contractions over the 4x4x4x3 coarse grid, then do the
//   trilinear image gather + nearest-neighbor label gather. Memory-bound; all data fits in L2.

typedef float v2f __attribute__((ext_vector_type(2)));
typedef float v8f __attribute__((ext_vector_type(8)));

#define NDIM 128
#define ALPHA 35.0f

__device__ __forceinline__ int clampi(int v, int lo, int hi) {
    return v < lo ? lo : (v > hi ? hi : v);
}

__device__ __forceinline__ float4 lerp4(const float4 a, const float4 b, const float t) {
    const float u = 1.0f - t;           // matches reference a*(1-t)+b*t
    float4 r;
    r.x = a.x * u + b.x * t;
    r.y = a.y * u + b.y * t;
    r.z = a.z * u + b.z * t;
    r.w = a.w * u + b.w * t;
    return r;
}

// ---------------------------------------------------------------------------
// Setup kernel: Wtab[d][i] = sum_t g[t] * CubicW[d+t-7][i], one float4 row per d.
// CubicW = jax.image.resize 'cubic' weights (Keys a=-0.5, per-output normalized).
// Gaussian: 16 taps at offsets -7..+8, sigma=2.5, normalized; SAME pad_lo=7.
// ---------------------------------------------------------------------------
__global__ __launch_bounds__(128) void make_wtab_kernel(float4* __restrict__ wtab) {
    __shared__ float wc[NDIM][4];
    const int t = threadIdx.x;                       // output index d in 0..127

    // cubic weights for output position t
    const float sf = (t + 0.5f) * (4.0f / 128.0f) - 0.5f;   // sample position in input coords
    float w[4];
    float sum = 0.0f;
#pragma unroll
    for (int i = 0; i < 4; ++i) {
        const float x = fabsf(sf - (float)i);
        float v;
        if (x < 1.0f)       v = ((1.5f * x - 2.5f) * x) * x + 1.0f;
        else if (x < 2.0f)  v = ((-0.5f * x + 2.5f) * x - 4.0f) * x + 2.0f;
        else                v = 0.0f;
        w[i] = v;
        sum += v;
    }
    const float inv_sum = 1.0f / sum;
#pragma unroll
    for (int i = 0; i < 4; ++i) wc[t][i] = w[i] * inv_sum;
    __syncthreads();

    // 16-tap Gaussian (offsets -7..+8), normalized
    float g[16];
    float gs = 0.0f;
#pragma unroll
    for (int o = 0; o < 16; ++o) {
        const float a = (float)o - 7.0f;
        g[o] = expf(-a * a / (2.0f * 2.5f * 2.5f));
        gs += g[o];
    }

    float acc[4] = {0.0f, 0.0f, 0.0f, 0.0f};
#pragma unroll
    for (int o = 0; o < 16; ++o) {
        const int s = t + o - 7;                     // zero padded outside [0,128)
        if (s >= 0 && s < NDIM) {
#pragma unroll
            for (int i = 0; i < 4; ++i) acc[i] += g[o] * wc[s][i];
        }
    }
    const float inv_gs = 1.0f / gs;
    wtab[t] = make_float4(acc[0] * inv_gs, acc[1] * inv_gs, acc[2] * inv_gs, acc[3] * inv_gs);
}

// ---------------------------------------------------------------------------
// Fused flow + warp kernel.
// Block: 256 threads = 8 wave32s. Block handles fixed (b, d) and 16 h values;
// wave wv handles w tile [16*wv, 16*wv+16). Each wave: 4 x V_WMMA_F32_16X16X4_F32.
// ---------------------------------------------------------------------------
__global__ __launch_bounds__(256) void elastic_warp_kernel(
    const float*  __restrict__ img,     // [2,128,128,128,4]
    const int*    __restrict__ lab,     // [2,128,128,128,1]
    const float*  __restrict__ coarse,  // [2,4,4,4,3]
    const float4* __restrict__ wtab,    // [128] composite weights
    float*        __restrict__ out)     // image floats then label floats
{
    __shared__ float cs[192];           // this batch's coarse grid
    __shared__ float Tl[8][16][17];     // per-wave T staging (padded stride 17 vs banks)

    const int b    = blockIdx.z;
    const int d    = blockIdx.y;
    const int h0   = blockIdx.x * 16;
    const int tid  = threadIdx.x;
    const int wv   = tid >> 5;          // wave id 0..7
    const int lane = tid & 31;
    const int half = lane >> 4;         // 0: lanes 0-15, 1: lanes 16-31
    const int lm   = lane & 15;
    const int w0   = wv * 16;
    const int kb   = half * 2;          // K base for this lane half (f32 WMMA layout)

    if (tid < 192) cs[tid] = coarse[b * 192 + tid];
    __builtin_prefetch(&wtab[d], 0, 3);             // -> global_prefetch_b8
    __syncthreads();

    // fd = W[d][0..3] (uniform across block)
    const float4 fdv = wtab[d];
    const float fdi[4] = {fdv.x, fdv.y, fdv.z, fdv.w};

    // ---- S = contract i:  S[j][k][c] = sum_i fd[i] * coarse[i][j][k][c]
    // Packed as B operand of WMMA#1: N = j*3+c (12 valid cols), K = k.
    v2f b1; b1.x = 0.0f; b1.y = 0.0f;
    if (lm < 12) {
        const int j = lm / 3;
        const int c = lm - j * 3;
        float s0 = 0.0f, s1 = 0.0f;
#pragma unroll
        for (int i = 0; i < 4; ++i) {
            s0 += fdi[i] * cs[i * 48 + j * 12 + kb * 3 + c];
            s1 += fdi[i] * cs[i * 48 + j * 12 + (kb + 1) * 3 + c];
        }
        b1.x = s0; b1.y = s1;
    }

    // ---- A operand of WMMA#1: rows = w (this wave's 16 w values), K = k
    const float4 wwv = wtab[w0 + lm];
    v2f a1;
    a1.x = half ? wwv.z : wwv.x;
    a1.y = half ? wwv.w : wwv.y;

    v8f czero = {};
    // T[w][j*3+c] = sum_k W[w][k] * S[j][k][c]
    v8f T = __builtin_amdgcn_wmma_f32_16x16x4_f32(false, a1, false, b1,
                                                  (short)0, czero, false, false);

    // Stage T through LDS to re-layout D->B for the second contraction.
#pragma unroll
    for (int r = 0; r < 8; ++r) Tl[wv][r + 8 * half][lm] = T[r];
    __syncthreads();

    // ---- A operand of WMMA#2: rows = h, K = j
    const float4 whv = wtab[h0 + lm];
    v2f a2;
    a2.x = half ? whv.z : whv.x;
    a2.y = half ? whv.w : whv.y;

    // flow_c[h][w] = sum_j W[h][j] * T[w][j*3+c]   (one WMMA per component)
    v8f Dc[3];
#pragma unroll
    for (int c = 0; c < 3; ++c) {
        v2f b2;
        b2.x = Tl[wv][lm][kb * 3 + c];        // B[K=kb,   N=w=lm]
        b2.y = Tl[wv][lm][(kb + 1) * 3 + c];  // B[K=kb+1, N=w=lm]
        Dc[c] = __builtin_amdgcn_wmma_f32_16x16x4_f32(false, a2, false, b2,
                                                      (short)0, czero, false, false);
    }

    // ---- Warp phase: lane l, reg r own voxel (h = h0 + r + 8*half, w = w0 + lm).
    const float4* img4   = (const float4*)img;     // C=4 channels per voxel
    const int     base_b = b * (NDIM * NDIM * NDIM);
    const int     w      = w0 + lm;
    float4*       outImg = (float4*)out;
    float*        outLab = out + (size_t)2 * NDIM * NDIM * NDIM * 4;

#pragma unroll
    for (int r = 0; r < 8; ++r) {
        const int h = h0 + r + 8 * half;
        const float wd = (float)d + ALPHA * Dc[0][r];
        const float wh = (float)h + ALPHA * Dc[1][r];
        const float ww = (float)w + ALPHA * Dc[2][r];

        const float fdd = floorf(wd), fhh = floorf(wh), fww = floorf(ww);
        const float td = wd - fdd, th = wh - fhh, tw = ww - fww;
        const int z0 = clampi((int)fdd, 0, NDIM - 1), z1 = clampi((int)fdd + 1, 0, NDIM - 1);
        const int y0 = clampi((int)fhh, 0, NDIM - 1), y1 = clampi((int)fhh + 1, 0, NDIM - 1);
        const int x0 = clampi((int)fww, 0, NDIM - 1), x1 = clampi((int)fww + 1, 0, NDIM - 1);

        auto at = [&](int z, int y, int x) {
            return img4[base_b + (z * NDIM + y) * NDIM + x];
        };
        const float4 c00 = lerp4(at(z0, y0, x0), at(z0, y0, x1), tw);
        const float4 c01 = lerp4(at(z0, y1, x0), at(z0, y1, x1), tw);
        const float4 c10 = lerp4(at(z1, y0, x0), at(z1, y0, x1), tw);
        const float4 c11 = lerp4(at(z1, y1, x0), at(z1, y1, x1), tw);
        const float4 cc0 = lerp4(c00, c01, th);
        const float4 cc1 = lerp4(c10, c11, th);
        const float4 res = lerp4(cc0, cc1, td);

        const int vi = ((b * NDIM + d) * NDIM + h) * NDIM + w;
        outImg[vi] = res;

        // nearest-neighbor label (round half-to-even == jnp.round)
        const int zn = clampi((int)rintf(wd), 0, NDIM - 1);
        const int yn = clampi((int)rintf(wh), 0, NDIM - 1);
        const int xn = clampi((int)rintf(ww), 0, NDIM - 1);
        outLab[vi] = (float)lab[base_b + (zn * NDIM + yn) * NDIM + xn];
    }
}

extern "C" void kernel_launch(void* const* d_in, const int* in_sizes, int n_in,
                              void* d_out, int out_size, void* d_ws, size_t ws_size,
                              hipStream_t stream) {
    const float* img    = (const float*)d_in[0];   // [2,128,128,128,4] f32
    const int*   lab    = (const int*)d_in[1];     // [2,128,128,128,1] i32
    const float* coarse = (const float*)d_in[2];   // [2,4,4,4,3] f32
    float4*      wtab   = (float4*)d_ws;           // 128 x float4 = 2 KB scratch
    float*       out    = (float*)d_out;

    hipLaunchKernelGGL(make_wtab_kernel, dim3(1), dim3(128), 0, stream, wtab);

    dim3 grid(NDIM / 16, NDIM, 2);                 // (h-tiles, d, b)
    hipLaunchKernelGGL(elastic_warp_kernel, grid, dim3(256), 0, stream,
                       img, lab, coarse, (const float4*)wtab, out);
}